// LATTE_75204877353792
// MI455X (gfx1250) — compile-verified
//
#include <hip/hip_runtime.h>
#include <hip/hip_bf16.h>

#define F_IN 256
#define D 128

typedef __attribute__((ext_vector_type(2))) float v2f;
typedef __attribute__((ext_vector_type(8))) float v8f;

// ---------- helpers ----------
// monotonic float -> uint key (exact order-preserving), for atomic segment-max
static __device__ __forceinline__ unsigned f32key(float f) {
  unsigned u = __builtin_bit_cast(unsigned, f);
  return (u & 0x80000000u) ? ~u : (u | 0x80000000u);
}
static __device__ __forceinline__ float keyf32(unsigned k) {
  unsigned u = (k & 0x80000000u) ? (k ^ 0x80000000u) : ~k;
  return __builtin_bit_cast(float, u);
}
static __device__ __forceinline__ float log_sigmoid(float x) {
  return fminf(x, 0.f) - log1pf(expf(-fabsf(x)));
}

// ---------- kernel 1: h = tanh(x @ W_lin^T) via f32 WMMA (16x16x4) ----------
// block = 256 threads = 8 waves. Wave w owns output cols [16w, 16w+16).
// Each wave computes TWO 16x16 M-tiles (rows r0..r0+31) sharing one B fragment:
// per K-chunk of 4: 3x global_load_b64 + 2x v_wmma_f32_16x16x4_f32 (independent
// accumulator chains c0/c1 -> no back-to-back D->C stall, half the B traffic).
//
// f32 WMMA fragment layout (ISA 7.12.2):
//   A 16x4 : lane L -> row M=L%16, holds K = 2*(L>=16), 2*(L>=16)+1  (float2)
//   B 4x16 : lane L -> col N=L%16, same K pair                        (float2)
//   C/D    : elem i of lane L -> (M = i + 8*(L>=16), N = L%16)
__global__ void __launch_bounds__(256)
k_gemm_tanh(const float* __restrict__ x, const float* __restrict__ Wlin,
            float* __restrict__ h, int N) {
  const int wave = threadIdx.x >> 5;
  const int lane = threadIdx.x & 31;
  const int half = lane >> 4;
  const int mrow = lane & 15;
  const int r0   = blockIdx.x * 32;

  int rowA0 = r0 + mrow;        if (rowA0 > N - 1) rowA0 = N - 1;  // clamp: EXEC must stay full
  int rowA1 = r0 + 16 + mrow;   if (rowA1 > N - 1) rowA1 = N - 1;
  const float* arow0 = x    + (size_t)rowA0 * F_IN + 2 * half;
  const float* arow1 = x    + (size_t)rowA1 * F_IN + 2 * half;
  const float* brow  = Wlin + (size_t)(wave * 16 + mrow) * F_IN + 2 * half;

  v8f c0 = {}, c1 = {};
  #pragma unroll
  for (int kc = 0; kc < F_IN / 4; ++kc) {
    const int k = kc * 4;
    v2f a0 = *reinterpret_cast<const v2f*>(arow0 + k);
    v2f a1 = *reinterpret_cast<const v2f*>(arow1 + k);
    v2f b  = *reinterpret_cast<const v2f*>(brow + k);
    c0 = __builtin_amdgcn_wmma_f32_16x16x4_f32(false, a0, false, b, (short)0, c0, false, false);
    c1 = __builtin_amdgcn_wmma_f32_16x16x4_f32(false, a1, false, b, (short)0, c1, false, false);
  }

  const int col = wave * 16 + mrow;
  #pragma unroll
  for (int i = 0; i < 8; ++i) {
    const int row = r0 + i + half * 8;
    if (row < N)      h[(size_t)row * D + col]        = tanhf(c0[i]);
    if (row + 16 < N) h[(size_t)(row + 16) * D + col] = tanhf(c1[i]);
  }
}

// ---------- kernel: zero/init scratch + loss ----------
__global__ void k_init(float* __restrict__ agg, float* __restrict__ denom,
                       unsigned* __restrict__ amax, float* __restrict__ loss, int N) {
  size_t idx = (size_t)blockIdx.x * 256 + threadIdx.x;
  if (idx < (size_t)N * D) agg[idx] = 0.f;
  if (idx < (size_t)N) { denom[idx] = 0.f; amax[idx] = 0u; }
  if (idx == 0) *loss = 0.f;
}

// ---------- kernel: score[n] = dot(h[n,:], attn_l) (one wave per node) ----------
__global__ void k_score(const float* __restrict__ h, const float* __restrict__ attn,
                        float* __restrict__ score, int N) {
  int t = blockIdx.x * 256 + threadIdx.x;
  int node = t >> 5, lane = t & 31;
  if (node >= N) return;
  float4 hv = reinterpret_cast<const float4*>(h)[(size_t)node * (D / 4) + lane];
  float4 av = reinterpret_cast<const float4*>(attn)[lane];
  float v = hv.x*av.x + hv.y*av.y + hv.z*av.z + hv.w*av.w;
  #pragma unroll
  for (int off = 16; off > 0; off >>= 1) v += __shfl_down(v, off, 32);
  if (lane == 0) score[node] = v;
}

// ---------- kernel: beta[n,:] = softmax(x[n,:] @ W_conv^T + b) (wave per node) ----------
__global__ void k_beta(const float* __restrict__ x, const float* __restrict__ Wc,
                       const float* __restrict__ bc, float* __restrict__ beta, int N) {
  int t = blockIdx.x * 256 + threadIdx.x;
  int node = t >> 5, lane = t & 31;
  if (node >= N) return;
  const float4* xr = reinterpret_cast<const float4*>(x + (size_t)node * F_IN);
  const float4* w0 = reinterpret_cast<const float4*>(Wc);
  const float4* w1 = reinterpret_cast<const float4*>(Wc + F_IN);
  float d0 = 0.f, d1 = 0.f;
  #pragma unroll
  for (int j = 0; j < 2; ++j) {
    float4 xv = xr[lane * 2 + j];
    float4 a = w0[lane * 2 + j], b = w1[lane * 2 + j];
    d0 += xv.x*a.x + xv.y*a.y + xv.z*a.z + xv.w*a.w;
    d1 += xv.x*b.x + xv.y*b.y + xv.z*b.z + xv.w*b.w;
  }
  #pragma unroll
  for (int off = 16; off > 0; off >>= 1) {
    d0 += __shfl_down(d0, off, 32);
    d1 += __shfl_down(d1, off, 32);
  }
  if (lane == 0) {
    float s0 = d0 + bc[0], s1 = d1 + bc[1];
    float m = fmaxf(s0, s1);
    float e0 = expf(s0 - m), e1 = expf(s1 - m);
    float inv = 1.f / (e0 + e1);
    beta[node * 2 + 0] = e0 * inv;
    beta[node * 2 + 1] = e1 * inv;
  }
}

// ---------- edge pass A: alpha + segment max (atomic on monotonic key) ----------
__global__ void k_edge_alpha(const int* __restrict__ ei, const float* __restrict__ score,
                             float* __restrict__ alpha, unsigned* __restrict__ amax, int E) {
  int e = blockIdx.x * 256 + threadIdx.x;
  if (e >= E) return;
  int s = ei[e], d = ei[E + e];
  float a = score[s] + score[d];
  alpha[e] = a;
  atomicMax(&amax[s], f32key(a));
}

// ---------- edge pass B: a = exp(alpha - amax[src]); denom += a ----------
__global__ void k_edge_exp(const int* __restrict__ ei, const float* __restrict__ alpha,
                           const unsigned* __restrict__ amax, float* __restrict__ aexp,
                           float* __restrict__ denom, int E) {
  int e = blockIdx.x * 256 + threadIdx.x;
  if (e >= E) return;
  int s = ei[e];
  float a = expf(alpha[e] - keyf32(amax[s]));
  aexp[e] = a;
  atomicAdd(&denom[s], a);
}

// ---------- edge pass C: agg[src,:] += h[dst,:] * w  (one wave per edge, float4) ----------
__global__ void k_edge_scatter(const int* __restrict__ ei, const float* __restrict__ aexp,
                               const float* __restrict__ denom, const float* __restrict__ h,
                               float* __restrict__ agg, int E) {
  int t = blockIdx.x * 256 + threadIdx.x;
  int e = t >> 5, lane = t & 31;
  if (e >= E) return;
  int s = ei[e], d = ei[E + e];
  float w = aexp[e] / (denom[s] + 1e-16f);
  float4 hv = reinterpret_cast<const float4*>(h)[(size_t)d * (D / 4) + lane];
  float* dst = agg + (size_t)s * D + lane * 4;
  atomicAdd(dst + 0, hv.x * w);
  atomicAdd(dst + 1, hv.y * w);
  atomicAdd(dst + 2, hv.z * w);
  atomicAdd(dst + 3, hv.w * w);
}

// ---------- out = beta0*agg + beta1*h ----------
__global__ void k_combine(const float* __restrict__ agg, const float* __restrict__ h,
                          const float* __restrict__ beta, float* __restrict__ out, int N) {
  size_t idx = (size_t)blockIdx.x * 256 + threadIdx.x;
  if (idx >= (size_t)N * D) return;
  int node = (int)(idx >> 7);
  out[idx] = beta[node * 2 + 0] * agg[idx] + beta[node * 2 + 1] * h[idx];
}

// ---------- loss = -sum logsig(e_pos) - sum logsig(-e_neg) ----------
__global__ void k_loss(const int* __restrict__ ei, const int* __restrict__ nei,
                       const float* __restrict__ score, float* __restrict__ loss, int E) {
  int e = blockIdx.x * 256 + threadIdx.x;
  float v = 0.f;
  if (e < E) {
    float ep = score[ei[e]]  + score[ei[E + e]];
    float en = score[nei[e]] + score[nei[E + e]];
    v = -(log_sigmoid(ep) + log_sigmoid(-en));
  }
  #pragma unroll
  for (int off = 16; off > 0; off >>= 1) v += __shfl_down(v, off, 32);
  if ((threadIdx.x & 31) == 0) atomicAdd(loss, v);
}

extern "C" void kernel_launch(void* const* d_in, const int* in_sizes, int n_in,
                              void* d_out, int out_size, void* d_ws, size_t ws_size,
                              hipStream_t stream) {
  const float* x     = (const float*)d_in[0];
  const float* Wlin  = (const float*)d_in[1];
  const float* Wconv = (const float*)d_in[2];
  const float* bconv = (const float*)d_in[3];
  const float* attn  = (const float*)d_in[4];
  const int*   edge  = (const int*)d_in[6];
  const int*   nedge = (const int*)d_in[7];
  const int N = in_sizes[5];        // x_index count == N_NODES
  const int E = in_sizes[6] / 2;    // edge_index is [2, E]

  // workspace carve-out (256B-aligned slices)
  char* ws = (char*)d_ws;
  size_t off = 0;
  auto carve = [&](size_t bytes) -> void* {
    void* p = ws + off;
    off = (off + bytes + 255) & ~(size_t)255;
    return p;
  };
  float*    h     = (float*)   carve((size_t)N * D * sizeof(float));
  float*    agg   = (float*)   carve((size_t)N * D * sizeof(float));
  float*    score = (float*)   carve((size_t)N * sizeof(float));
  float*    beta  = (float*)   carve((size_t)N * 2 * sizeof(float));
  unsigned* amax  = (unsigned*)carve((size_t)N * sizeof(unsigned));
  float*    denom = (float*)   carve((size_t)N * sizeof(float));
  float*    alpha = (float*)   carve((size_t)E * sizeof(float));
  float*    aexp  = (float*)   carve((size_t)E * sizeof(float));

  float* out  = (float*)d_out;
  float* loss = out + (size_t)N * D;   // tuple: (out [N,D], loss scalar)

  const int ND = N * D;
  k_init        <<<(ND + 255) / 256,      256, 0, stream>>>(agg, denom, amax, loss, N);
  k_gemm_tanh   <<<(N + 31) / 32,         256, 0, stream>>>(x, Wlin, h, N);
  k_score       <<<(N * 32 + 255) / 256,  256, 0, stream>>>(h, attn, score, N);
  k_beta        <<<(N * 32 + 255) / 256,  256, 0, stream>>>(x, Wconv, bconv, beta, N);
  k_edge_alpha  <<<(E + 255) / 256,       256, 0, stream>>>(edge, score, alpha, amax, E);
  k_edge_exp    <<<(E + 255) / 256,       256, 0, stream>>>(edge, alpha, amax, aexp, denom, E);
  k_edge_scatter<<<(E * 32 + 255) / 256,  256, 0, stream>>>(edge, aexp, denom, h, agg, E);
  k_combine     <<<(ND + 255) / 256,      256, 0, stream>>>(agg, h, beta, out, N);
  k_loss        <<<(E + 255) / 256,       256, 0, stream>>>(edge, nedge, score, loss, E);
}